// MultiHeadAttention_70866960384272
// MI455X (gfx1250) — compile-verified
//
#include <hip/hip_runtime.h>
#include <math.h>

// ---------------------------------------------------------------------------
// MHA forward for MI455X (gfx1250): fp32 V_WMMA_F32_16X16X4_F32 matmuls,
// double-buffered GLOBAL_LOAD_ASYNC_TO_LDS_B128 staging, flash attention.
// ---------------------------------------------------------------------------

typedef __attribute__((ext_vector_type(2))) float v2f;
typedef __attribute__((ext_vector_type(8))) float v8f;

#define DMODEL 1024
#define NHEAD  16
#define HD     64
#define SEQ    2048
#define BATCH  2
#define NROWS  (BATCH * SEQ)   // 4096

__device__ __forceinline__ v8f wmma_f32(v2f a, v2f b, v8f c) {
  // 8 args: (neg_a, A, neg_b, B, c_mod, C, reuse_a, reuse_b)
  return __builtin_amdgcn_wmma_f32_16x16x4_f32(false, a, false, b, (short)0, c,
                                               false, false);
}

// Low 32 bits of a flat LDS pointer == raw LDS byte offset (ISA aperture rule)
__device__ __forceinline__ unsigned lds_lo(const void* p) {
  return (unsigned)(uintptr_t)p;
}

// 16B global -> LDS async copy (ASYNCcnt-tracked, no VGPR staging)
__device__ __forceinline__ void async_b128(unsigned lds_addr, const float* g) {
  asm volatile("global_load_async_to_lds_b128 %0, %1, off"
               :: "v"(lds_addr), "v"(g)
               : "memory");
}
__device__ __forceinline__ void wait_async0() {
  asm volatile("s_wait_asynccnt 0x0" ::: "memory");
}

// ---------------------------------------------------------------------------
// Tiled GEMM:  C[NROWS x NN] = A[NROWS x 1024] * Bmat[1024 x NN] + bias
// MODE 0: scatter columns into per-head Q/K/V buffers [B,H,S,hd]
// MODE 1: plain row-major store to out0
// 256 threads (8 waves), BM=64 BN=128 BK=32, wave grid 2x4, 2x2 WMMA tiles.
// LDS double-buffered, filled by async B128 copies.
// ---------------------------------------------------------------------------
template <int NN, int MODE>
__global__ __launch_bounds__(256) void gemm_kernel(
    const float* __restrict__ Amat, const float* __restrict__ Bmat,
    const float* __restrict__ bias, float* __restrict__ out0,
    float* __restrict__ out1, float* __restrict__ out2) {
  constexpr int BM = 64, BN = 128, BK = 32;
  constexpr int AP = BK + 4;    // 36 floats = 144B pitch (16B multiple)
  constexpr int BP = BN + 16;   // 144 floats = 576B pitch (bank-split halves)
  __shared__ alignas(16) float As[2][BM][AP];   // natural [m][k]
  __shared__ alignas(16) float Bs[2][BK][BP];   // natural [k][n]

  const int tid  = threadIdx.x;
  const int lane = tid & 31;
  const int half = lane >> 4;
  const int l16  = lane & 15;
  const int wave = tid >> 5;
  const int wm   = wave >> 2;   // 0..1
  const int wn   = wave & 3;    // 0..3

  const int M0 = blockIdx.y * BM;
  const int N0 = blockIdx.x * BN;

  auto stage = [&](int kb, int buf) {
#pragma unroll
    for (int t = 0; t < 2; ++t) {          // A: 64 rows x 8 x 16B
      int c = tid + t * 256;
      int row = c >> 3, c16 = c & 7;
      async_b128(lds_lo(&As[buf][row][c16 * 4]),
                 Amat + (size_t)(M0 + row) * DMODEL + kb + c16 * 4);
    }
#pragma unroll
    for (int t = 0; t < 4; ++t) {          // B: 32 rows x 32 x 16B
      int c = tid + t * 256;
      int row = c >> 5, c16 = c & 31;
      async_b128(lds_lo(&Bs[buf][row][c16 * 4]),
                 Bmat + (size_t)(kb + row) * NN + N0 + c16 * 4);
    }
  };

  v8f acc[2][2] = {};
  stage(0, 0);

  for (int it = 0; it < DMODEL / BK; ++it) {
    wait_async0();
    __syncthreads();
    if (it + 1 < DMODEL / BK) stage((it + 1) * BK, (it + 1) & 1);
    const int buf = it & 1;

#pragma unroll
    for (int k0 = 0; k0 < BK; k0 += 4) {
      v2f a[2], b[2];
#pragma unroll
      for (int tm = 0; tm < 2; ++tm) {
        int mr = wm * 32 + tm * 16 + l16;
        a[tm].x = As[buf][mr][k0 + 2 * half];      // K0/K2 per lane half
        a[tm].y = As[buf][mr][k0 + 2 * half + 1];  // K1/K3 per lane half
      }
#pragma unroll
      for (int tn = 0; tn < 2; ++tn) {
        int nc = wn * 32 + tn * 16 + l16;
        b[tn].x = Bs[buf][k0 + half][nc];
        b[tn].y = Bs[buf][k0 + 2 + half][nc];
      }
#pragma unroll
      for (int tm = 0; tm < 2; ++tm)
#pragma unroll
        for (int tn = 0; tn < 2; ++tn)
          acc[tm][tn] = wmma_f32(a[tm], b[tn], acc[tm][tn]);
    }
    // no trailing barrier needed: next iteration's wait+barrier orders reuse
  }

  // Epilogue: C/D layout -> VGPR i, lanes 0-15: M=i, lanes 16-31: M=i+8
#pragma unroll
  for (int tm = 0; tm < 2; ++tm)
#pragma unroll
    for (int tn = 0; tn < 2; ++tn)
#pragma unroll
      for (int i = 0; i < 8; ++i) {
        int gm = M0 + wm * 32 + tm * 16 + i + half * 8;
        int gn = N0 + wn * 32 + tn * 16 + l16;
        float v = acc[tm][tn][i] + bias[gn];
        if (MODE == 0) {
          int b_ = gm >> 11;
          int s_ = gm & (SEQ - 1);
          int h_ = gn / 192;
          int j  = gn - h_ * 192;
          size_t dst = (((size_t)b_ * NHEAD + h_) * SEQ + s_) * HD + (j & 63);
          if (j < 64)       out0[dst] = v;  // Q
          else if (j < 128) out1[dst] = v;  // K
          else              out2[dst] = v;  // V
        } else {
          out0[(size_t)gm * NN + gn] = v;
        }
      }
}

// ---------------------------------------------------------------------------
// Flash attention per (b, h, 64-query tile). 256 threads = 8 waves.
// Wave w owns q-block qb = w/2 (16 rows) and column half nb0 = (w&1)*32.
// K/V chunks double-buffered via async B128 copies; Q staged once.
// All LDS tiles natural row-major with bank-aware pads.
// ---------------------------------------------------------------------------
__global__ __launch_bounds__(256) void attn_kernel(
    const float* __restrict__ q, const float* __restrict__ k,
    const float* __restrict__ v, const float* __restrict__ mask,
    float* __restrict__ out) {
  __shared__ alignas(16) float Qs[64][68];       // [q][hd]
  __shared__ alignas(16) float Ks[2][64][68];    // [key][hd]
  __shared__ alignas(16) float Vs[2][64][80];    // [key][hd], pitch 64+16
  __shared__ float Ps[64][68];                   // [q][key] exp-scores
  __shared__ float alphas[64];
  __shared__ float lrow[64];

  const int qt = blockIdx.x & 31;       // S/64 tiles
  const int bh = blockIdx.x >> 5;
  const int b_ = bh >> 4;
  const int h_ = bh & 15;
  const int q0 = qt * 64;
  const float scale = 0.125f;           // 1/sqrt(64)

  const size_t head_off = (size_t)bh * SEQ * HD;

  const int tid  = threadIdx.x;
  const int lane = tid & 31;
  const int half = lane >> 4;
  const int l16  = lane & 15;
  const int wave = tid >> 5;
  const int qb   = wave >> 1;           // 0..3
  const int nb0  = (wave & 1) * 32;

  auto stage_kv = [&](int j0, int buf) {
#pragma unroll
    for (int t = 0; t < 4; ++t) {          // K: 64 rows x 16 x 16B
      int c = tid + t * 256;
      int row = c >> 4, c16 = c & 15;
      async_b128(lds_lo(&Ks[buf][row][c16 * 4]),
                 k + head_off + (size_t)(j0 + row) * HD + c16 * 4);
    }
#pragma unroll
    for (int t = 0; t < 4; ++t) {          // V: 64 rows x 16 x 16B
      int c = tid + t * 256;
      int row = c >> 4, c16 = c & 15;
      async_b128(lds_lo(&Vs[buf][row][c16 * 4]),
                 v + head_off + (size_t)(j0 + row) * HD + c16 * 4);
    }
  };

  // Stage Q once + first K/V chunk
#pragma unroll
  for (int t = 0; t < 4; ++t) {
    int c = tid + t * 256;
    int row = c >> 4, c16 = c & 15;
    async_b128(lds_lo(&Qs[row][c16 * 4]),
               q + head_off + (size_t)(q0 + row) * HD + c16 * 4);
  }
  stage_kv(0, 0);

  float m_r = -1e30f, l_r = 0.0f;       // per-row state, valid for tid<64
  v8f o_acc[2] = {};

  for (int j = 0; j < SEQ / 64; ++j) {
    wait_async0();
    __syncthreads();
    if (j + 1 < SEQ / 64) stage_kv((j + 1) * 64, (j + 1) & 1);
    const int buf = j & 1;
    const int j0 = j * 64;

    // Scores: S = Q @ K^T * scale + mask  (two 16x16 tiles per wave)
#pragma unroll
    for (int tn = 0; tn < 2; ++tn) {
      v8f sacc = {};
#pragma unroll
      for (int k0 = 0; k0 < HD; k0 += 4) {
        v2f a, bb;
        int mr = qb * 16 + l16;
        a.x = Qs[mr][k0 + 2 * half];
        a.y = Qs[mr][k0 + 2 * half + 1];
        int nc = nb0 + tn * 16 + l16;
        bb.x = Ks[buf][nc][k0 + half];       // B[k][n] = K[n][k]
        bb.y = Ks[buf][nc][k0 + 2 + half];
        sacc = wmma_f32(a, bb, sacc);
      }
#pragma unroll
      for (int i = 0; i < 8; ++i) {
        int qq = qb * 16 + i + half * 8;
        int kk = nb0 + tn * 16 + l16;
        float sv = sacc[i] * scale +
                   mask[((size_t)b_ * SEQ + (q0 + qq)) * SEQ + j0 + kk];
        Ps[qq][kk] = sv;                     // natural store (lane-contiguous)
      }
    }
    __syncthreads();

    // Online softmax: lane t < 64 owns query row t (contiguous row scan)
    if (tid < 64) {
      float mc = -1e30f;
      for (int kk = 0; kk < 64; ++kk) mc = fmaxf(mc, Ps[tid][kk]);
      float mn = fmaxf(m_r, mc);
      float al = __expf(m_r - mn);
      float sum = 0.0f;
      for (int kk = 0; kk < 64; ++kk) {
        float e = __expf(Ps[tid][kk] - mn);
        Ps[tid][kk] = e;
        sum += e;
      }
      l_r = l_r * al + sum;
      m_r = mn;
      alphas[tid] = al;
    }
    __syncthreads();

    // Rescale running O, then O += P @ V
#pragma unroll
    for (int tn = 0; tn < 2; ++tn)
#pragma unroll
      for (int i = 0; i < 8; ++i) {
        int qq = qb * 16 + i + half * 8;
        o_acc[tn][i] *= alphas[qq];
      }
#pragma unroll
    for (int k0 = 0; k0 < 64; k0 += 4) {
      v2f a;
      int mr = qb * 16 + l16;
      a.x = Ps[mr][k0 + 2 * half];
      a.y = Ps[mr][k0 + 2 * half + 1];
#pragma unroll
      for (int tn = 0; tn < 2; ++tn) {
        v2f bb;
        int nc = nb0 + tn * 16 + l16;
        bb.x = Vs[buf][k0 + half][nc];
        bb.y = Vs[buf][k0 + 2 + half][nc];
        o_acc[tn] = wmma_f32(a, bb, o_acc[tn]);
      }
    }
    // next iteration's wait+barrier orders all LDS reuse
  }

  if (tid < 64) lrow[tid] = l_r;
  __syncthreads();

  // Final normalize + store in the reference's flat reshape order:
  // out[b*S*D + h*S*HD + s*HD + dd]
#pragma unroll
  for (int tn = 0; tn < 2; ++tn)
#pragma unroll
    for (int i = 0; i < 8; ++i) {
      int qq = qb * 16 + i + half * 8;
      int dd = nb0 + tn * 16 + l16;
      float val = o_acc[tn][i] / lrow[qq];
      out[(size_t)b_ * SEQ * DMODEL + (size_t)h_ * SEQ * HD +
          (size_t)(q0 + qq) * HD + dd] = val;
    }
}

// ---------------------------------------------------------------------------
extern "C" void kernel_launch(void* const* d_in, const int* in_sizes, int n_in,
                              void* d_out, int out_size, void* d_ws,
                              size_t ws_size, hipStream_t stream) {
  const float* x    = (const float*)d_in[0];
  const float* mask = (const float*)d_in[1];
  const float* Wqkv = (const float*)d_in[2];
  const float* bqkv = (const float*)d_in[3];
  const float* Wo   = (const float*)d_in[4];
  const float* bo   = (const float*)d_in[5];
  float* out = (float*)d_out;

  // Workspace: Q,K,V [B,H,S,hd] + attention output, 4 x 16 MB
  const size_t HBUF = (size_t)BATCH * NHEAD * SEQ * HD;  // 4,194,304
  float* ws   = (float*)d_ws;
  float* qbuf = ws;
  float* kbuf = ws + HBUF;
  float* vbuf = ws + 2 * HBUF;
  float* obuf = ws + 3 * HBUF;

  dim3 blk(256);

  // 1) QKV projection + bias, scattered to per-head Q/K/V
  gemm_kernel<3 * DMODEL, 0>
      <<<dim3((3 * DMODEL) / 128, NROWS / 64), blk, 0, stream>>>(
          x, Wqkv, bqkv, qbuf, kbuf, vbuf);

  // 2) Flash attention per (b, h, q-tile)
  attn_kernel<<<dim3(BATCH * NHEAD * (SEQ / 64)), blk, 0, stream>>>(
      qbuf, kbuf, vbuf, mask, obuf);

  // 3) Output projection + bias
  gemm_kernel<DMODEL, 1><<<dim3(DMODEL / 128, NROWS / 64), blk, 0, stream>>>(
      obuf, Wo, bo, out, nullptr, nullptr);
}